// LinkPredictor_86414741996077
// MI455X (gfx1250) — compile-verified
//
#include <hip/hip_runtime.h>

#define D 128

typedef __attribute__((ext_vector_type(2))) float v2f;
typedef __attribute__((ext_vector_type(8))) float v8f;

// ---------------- utility fills ----------------
__global__ void k_fill(float* __restrict__ p, float v, size_t n) {
  size_t i = (size_t)blockIdx.x * blockDim.x + threadIdx.x;
  size_t st = (size_t)gridDim.x * blockDim.x;
  for (; i < n; i += st) p[i] = v;
}

// ---------------- degree accumulation (self-loop folded in via init=1.0) ----
__global__ void k_deg(const long long* __restrict__ dst, float* __restrict__ deg, int E) {
  int i = blockIdx.x * blockDim.x + threadIdx.x;
  int st = gridDim.x * blockDim.x;
  for (; i < E; i += st) {
    unsafeAtomicAdd(&deg[(int)dst[i]], 1.0f);
  }
}

__global__ void k_rsqrt(float* __restrict__ d, int n) {
  int i = blockIdx.x * blockDim.x + threadIdx.x;
  if (i < n) d[i] = rsqrtf(d[i]);
}

// ---------------- fp32 WMMA GEMM: out[M x 128] = A[M x 128] * W[128 x 128] --
// block = 256 threads = 8 waves; each wave owns one 16-col N tile; block owns
// one 16-row M tile. K-loop: 32 x v_wmma_f32_16x16x4_f32.
__global__ void k_gemm_wmma(const float* __restrict__ A, const float* __restrict__ W,
                            float* __restrict__ out) {
  const int wave = threadIdx.x >> 5;     // 0..7 -> N tile
  const int lane = threadIdx.x & 31;
  const int lm   = lane & 15;
  const int hi   = lane >> 4;            // 0: K pair {0,1} / M 0..7 ; 1: K pair {2,3} / M 8..15
  const int m0   = blockIdx.x * 16;
  const int n    = wave * 16 + lm;

  const float2* arow = (const float2*)(A + (size_t)(m0 + lm) * D);
  v8f c = {};
  for (int kb = 0; kb < D; kb += 4) {
    const int ka = kb + 2 * hi;          // even -> 8B aligned
    float2 av = arow[ka >> 1];
    v2f a; a.x = av.x; a.y = av.y;       // VGPR0 = K+0(/+2), VGPR1 = K+1(/+3)
    v2f b;
    b.x = W[(size_t)ka * D + n];
    b.y = W[(size_t)(ka + 1) * D + n];
    c = __builtin_amdgcn_wmma_f32_16x16x4_f32(false, a, false, b, (short)0, c,
                                              false, false);
  }
  // C/D layout: VGPR r -> row m0 + 8*hi + r, col n
  float* orow = out + (size_t)(m0 + 8 * hi) * D + n;
#pragma unroll
  for (int r = 0; r < 8; ++r) orow[(size_t)r * D] = c[r];
}

// ---------------- edge scatter: acc[dst] += t[src] * dinv[src]*dinv[dst] ----
// one wave32 per edge; lane loads float4 -> 128 floats per row; L2-resident.
__global__ void k_scatter(const float* __restrict__ t, const long long* __restrict__ src,
                          const long long* __restrict__ dst, const float* __restrict__ dinv,
                          float* __restrict__ acc, int E) {
  const int lane = threadIdx.x & 31;
  int w  = (blockIdx.x * blockDim.x + threadIdx.x) >> 5;
  int nw = (gridDim.x * blockDim.x) >> 5;
  for (int e = w; e < E; e += nw) {
    const int s = (int)src[e];
    const int d = (int)dst[e];
    const float norm = dinv[s] * dinv[d];
    float4 hv = ((const float4*)(t + (size_t)s * D))[lane];
    float* orow = acc + (size_t)d * D + lane * 4;
    unsafeAtomicAdd(orow + 0, hv.x * norm);
    unsafeAtomicAdd(orow + 1, hv.y * norm);
    unsafeAtomicAdd(orow + 2, hv.z * norm);
    unsafeAtomicAdd(orow + 3, hv.w * norm);
  }
}

// -------- epilogue: out = [relu](acc + t*dinv^2 (self loop) + bias) ---------
__global__ void k_finish(const float* __restrict__ acc, const float* __restrict__ t,
                         const float* __restrict__ dinv, const float* __restrict__ bias,
                         float* __restrict__ out, size_t nd, int do_relu) {
  size_t i = (size_t)blockIdx.x * blockDim.x + threadIdx.x;
  size_t st = (size_t)gridDim.x * blockDim.x;
  for (; i < nd; i += st) {
    const size_t node = i >> 7;
    const int c = (int)(i & (D - 1));
    const float dv = dinv[node];
    float v = acc[i] + t[i] * dv * dv + bias[c];
    if (do_relu) v = fmaxf(v, 0.0f);
    out[i] = v;
  }
}

// ---------------- decode: score[e] = dot(z[src], z[dst]) --------------------
__global__ void k_decode(const float* __restrict__ z, const long long* __restrict__ s,
                         const long long* __restrict__ d, float* __restrict__ out, int E) {
  const int lane = threadIdx.x & 31;
  int w  = (blockIdx.x * blockDim.x + threadIdx.x) >> 5;
  int nw = (gridDim.x * blockDim.x) >> 5;
  for (int e = w; e < E; e += nw) {
    const int si = (int)s[e];
    const int di = (int)d[e];
    float4 av = ((const float4*)(z + (size_t)si * D))[lane];
    float4 bv = ((const float4*)(z + (size_t)di * D))[lane];
    float p = av.x * bv.x + av.y * bv.y + av.z * bv.z + av.w * bv.w;
#pragma unroll
    for (int off = 16; off > 0; off >>= 1) p += __shfl_xor(p, off, 32);
    if (lane == 0) out[e] = p;
  }
}

extern "C" void kernel_launch(void* const* d_in, const int* in_sizes, int n_in,
                              void* d_out, int out_size, void* d_ws, size_t ws_size,
                              hipStream_t stream) {
  const float*     x    = (const float*)d_in[0];
  const float*     W1   = (const float*)d_in[1];
  const float*     b1   = (const float*)d_in[2];
  const float*     W2   = (const float*)d_in[3];
  const float*     b2   = (const float*)d_in[4];
  const long long* eidx = (const long long*)d_in[5];
  const long long* pos  = (const long long*)d_in[6];
  const long long* neg  = (const long long*)d_in[7];

  const int N  = in_sizes[0] / D;   // 100000
  const int E  = in_sizes[5] / 2;   // 3200000
  const int Ep = in_sizes[6] / 2;   // 100000
  const int En = in_sizes[7] / 2;   // 100000
  const size_t ND = (size_t)N * D;

  // workspace layout
  float* dinv = (float*)d_ws;
  float* t    = dinv + (((size_t)N + 127) & ~(size_t)127); // GEMM output (pre-aggregation)
  float* a    = t + ND;                                    // aggregation accumulator / h
  float* outp = (float*)d_out;                             // pos scores
  float* outn = outp + Ep;                                 // neg scores
  float* zreg = outn + En;                                 // z [N x 128]

  const long long* esrc = eidx;
  const long long* edst = eidx + E;

  // degrees (init 1.0 = self loop), then dinv = rsqrt(deg)
  k_fill<<<512, 256, 0, stream>>>(dinv, 1.0f, (size_t)N);
  k_deg<<<2048, 256, 0, stream>>>(edst, dinv, E);
  k_rsqrt<<<(N + 255) / 256, 256, 0, stream>>>(dinv, N);

  // ---- layer 1: t = x @ W1 ; a = scatter(t) ; a = relu(a + t*dinv^2 + b1)
  k_gemm_wmma<<<N / 16, 256, 0, stream>>>(x, W1, t);
  k_fill<<<4096, 256, 0, stream>>>(a, 0.0f, ND);
  k_scatter<<<4096, 256, 0, stream>>>(t, esrc, edst, dinv, a, E);
  k_finish<<<16384, 256, 0, stream>>>(a, t, dinv, b1, a, ND, 1);

  // ---- layer 2: t = a @ W2 ; z = scatter(t) ; z = z + t*dinv^2 + b2
  k_gemm_wmma<<<N / 16, 256, 0, stream>>>(a, W2, t);
  k_fill<<<4096, 256, 0, stream>>>(zreg, 0.0f, ND);
  k_scatter<<<4096, 256, 0, stream>>>(t, esrc, edst, dinv, zreg, E);
  k_finish<<<16384, 256, 0, stream>>>(zreg, t, dinv, b2, zreg, ND, 0);

  // ---- decode
  k_decode<<<(Ep + 7) / 8, 256, 0, stream>>>(zreg, pos, pos + Ep, outp, Ep);
  k_decode<<<(En + 7) / 8, 256, 0, stream>>>(zreg, neg, neg + En, outn, En);
}